// GCNLinkPredictor_83313775608468
// MI455X (gfx1250) — compile-verified
//
#include <hip/hip_runtime.h>

// Problem sizes (fixed by the reference)
#define N_NODES 50000
#define N_EDGES 800000
#define N_PAIRS 400000
#define DIM     128
#define LDS_STRIDE 132   // 132 % 4 == 0 (float4 stores OK); 4*row+k banks distinct

typedef float v2f __attribute__((ext_vector_type(2)));
typedef float v8f __attribute__((ext_vector_type(8)));

// ---------------------------------------------------------------- degree prep
__global__ void k_init_deg(float* __restrict__ deg) {
    int i = blockIdx.x * blockDim.x + threadIdx.x;
    if (i < N_NODES) deg[i] = 1.0f;               // self-loop
}

__global__ void k_scatter_deg(const int* __restrict__ dst, float* __restrict__ deg) {
    int e = blockIdx.x * blockDim.x + threadIdx.x;
    if (e < N_EDGES) atomicAdd(&deg[dst[e]], 1.0f);
}

__global__ void k_rsqrt_inplace(float* __restrict__ deg) {
    int i = blockIdx.x * blockDim.x + threadIdx.x;
    if (i < N_NODES) deg[i] = rsqrtf(deg[i]);     // deg >= 1 always
}

// ------------------------------------------------- H = act(X [+bias]) @ W (WMMA)
// Block: 256 threads = 8 waves. Block handles 16 rows; wave w -> 16-col tile w.
// A (16x4 f32) per ISA layout: lane L -> M = L%16, VGPR0 = K = 2*(L/16), VGPR1 = K+1.
// B (4x16 f32) mirrored:       lane L -> N = L%16, VGPR0 = K = 2*(L/16), VGPR1 = K+1.
// C/D (16x16 f32): VGPR v, lane L -> M = v + 8*(L/16), N = L%16.
__launch_bounds__(256, 4)
__global__ void k_matmul_wmma(const float* __restrict__ X,
                              const float* __restrict__ W,
                              const float* __restrict__ bias,   // non-null => relu(x+bias)
                              float* __restrict__ H) {
    __shared__ float lds[16 * LDS_STRIDE];
    const int rowBase = blockIdx.x * 16;
    const int t = threadIdx.x;

    // Stage 16x128 X tile into LDS (2 float4 per thread), fusing bias+ReLU for layer 2.
    #pragma unroll
    for (int rep = 0; rep < 2; ++rep) {
        int idx = t + rep * 256;          // 0..511 float4 slots
        int r   = idx >> 5;               // 0..15
        int c4  = idx & 31;               // 0..31
        float4 v = *(const float4*)(X + (size_t)(rowBase + r) * DIM + c4 * 4);
        if (bias) {
            float4 b = *(const float4*)(bias + c4 * 4);
            v.x = fmaxf(v.x + b.x, 0.0f);
            v.y = fmaxf(v.y + b.y, 0.0f);
            v.z = fmaxf(v.z + b.z, 0.0f);
            v.w = fmaxf(v.w + b.w, 0.0f);
        }
        *(float4*)(&lds[r * LDS_STRIDE + c4 * 4]) = v;
    }
    __syncthreads();

    const int wave = t >> 5;
    const int lane = t & 31;
    const int half = lane >> 4;           // 0 or 1
    const int l16  = lane & 15;
    const int colBase = wave * 16;

    v8f acc = {0.f, 0.f, 0.f, 0.f, 0.f, 0.f, 0.f, 0.f};

    #pragma unroll
    for (int k0 = 0; k0 < DIM; k0 += 4) {
        const int kk = k0 + half * 2;
        float2 av = *(const float2*)(&lds[l16 * LDS_STRIDE + kk]);
        v2f a, b;
        a[0] = av.x;
        a[1] = av.y;
        b[0] = W[(size_t)kk * DIM + colBase + l16];
        b[1] = W[(size_t)(kk + 1) * DIM + colBase + l16];
        // D = A(16x4) * B(4x16) + C, full fp32
        acc = __builtin_amdgcn_wmma_f32_16x16x4_f32(false, a, false, b,
                                                    (short)0, acc, false, false);
    }

    float* outp = H + (size_t)rowBase * DIM + colBase;
    #pragma unroll
    for (int v = 0; v < 8; ++v) {
        outp[(size_t)(v + half * 8) * DIM + l16] = acc[v];
    }
}

// ------------------------------------- out[i] = h[i]*dinv[i]^2 (+ bias)  (self-loop)
__global__ void k_init_out(const float* __restrict__ h, const float* __restrict__ dinv,
                           const float* __restrict__ bias, float* __restrict__ out) {
    int idx = blockIdx.x * blockDim.x + threadIdx.x;   // [0, N*128)
    int i = idx >> 7;
    int f = idx & 127;
    float di = dinv[i];
    float v = h[idx] * di * di;
    if (bias) v += bias[f];
    out[idx] = v;
}

// --------------------------------- per edge: out[dst] += h[src] * dinv[s]*dinv[d]
// One wave32 per edge; each lane owns a float4 of the 128-float row.
__launch_bounds__(256)
__global__ void k_scatter_edges(const int* __restrict__ src, const int* __restrict__ dst,
                                const float* __restrict__ h, const float* __restrict__ dinv,
                                float* __restrict__ out) {
    const int wid  = threadIdx.x >> 5;
    const int lane = threadIdx.x & 31;
    const int e = blockIdx.x * 8 + wid;
    if (e >= N_EDGES) return;
    const int s = src[e];
    const int d = dst[e];
    const float norm = dinv[s] * dinv[d];
    const float4 hv = *(const float4*)(h + (size_t)s * DIM + lane * 4);
    float* op = out + (size_t)d * DIM + lane * 4;
    atomicAdd(op + 0, hv.x * norm);
    atomicAdd(op + 1, hv.y * norm);
    atomicAdd(op + 2, hv.z * norm);
    atomicAdd(op + 3, hv.w * norm);
}

// ------------------------------------------------ decode: dot(z[a], z[b]) per pair
__launch_bounds__(256)
__global__ void k_decode(const int* __restrict__ pa, const int* __restrict__ pb,
                         const float* __restrict__ z, float* __restrict__ out) {
    const int wid  = threadIdx.x >> 5;
    const int lane = threadIdx.x & 31;
    const int p = blockIdx.x * 8 + wid;
    if (p >= N_PAIRS) return;
    const int a = pa[p];
    const int b = pb[p];
    const float4 za = *(const float4*)(z + (size_t)a * DIM + lane * 4);
    const float4 zb = *(const float4*)(z + (size_t)b * DIM + lane * 4);
    float s = za.x * zb.x + za.y * zb.y + za.z * zb.z + za.w * zb.w;
    #pragma unroll
    for (int off = 16; off > 0; off >>= 1) s += __shfl_down(s, off);
    if (lane == 0) out[p] = s;
}

// -------------------------------------------------------------------- launcher
extern "C" void kernel_launch(void* const* d_in, const int* in_sizes, int n_in,
                              void* d_out, int out_size, void* d_ws, size_t ws_size,
                              hipStream_t stream) {
    const int*   edge_index = (const int*)d_in[0];   // [2, E] row-major
    const int*   edge_pairs = (const int*)d_in[1];   // [2, P]
    const float* emb        = (const float*)d_in[2]; // [N, 128]
    const float* W1         = (const float*)d_in[3]; // [128, 128]
    const float* b1         = (const float*)d_in[4]; // [128]
    const float* W2         = (const float*)d_in[5]; // [128, 128]
    const float* b2         = (const float*)d_in[6]; // [128]
    float*       out        = (float*)d_out;         // [P]

    const int* src = edge_index;            // row 0
    const int* dst = edge_index + N_EDGES;  // row 1
    const int* pa  = edge_pairs;
    const int* pb  = edge_pairs + N_PAIRS;

    // Workspace layout (~51.5 MB): dinv | h | agg
    char* ws = (char*)d_ws;
    size_t off = 0;
    float* dinv = (float*)(ws + off); off += ((size_t)N_NODES * 4 + 255) & ~(size_t)255;
    float* h    = (float*)(ws + off); off += (size_t)N_NODES * DIM * 4;
    float* agg  = (float*)(ws + off); off += (size_t)N_NODES * DIM * 4;
    (void)ws_size; (void)in_sizes; (void)n_in; (void)out_size;

    const int T = 256;
    const int gN    = (N_NODES + T - 1) / T;        // 196
    const int gE    = (N_EDGES + T - 1) / T;        // 3125
    const int gNF   = (N_NODES * DIM) / T;          // 25000 (exact)
    const int gMM   = N_NODES / 16;                 // 3125  (exact)
    const int gEw   = N_EDGES / 8;                  // 100000 (exact, wave/edge)
    const int gPw   = N_PAIRS / 8;                  // 50000  (exact, wave/pair)

    // deg -> dinv (in place)
    k_init_deg<<<gN, T, 0, stream>>>(dinv);
    k_scatter_deg<<<gE, T, 0, stream>>>(dst, dinv);
    k_rsqrt_inplace<<<gN, T, 0, stream>>>(dinv);

    // Layer 1: h1 = emb @ W1 ; agg1 = scatter(h1) (b1 + ReLU fused into layer-2 A-load)
    k_matmul_wmma<<<gMM, T, 0, stream>>>(emb, W1, nullptr, h);
    k_init_out<<<gNF, T, 0, stream>>>(h, dinv, nullptr, agg);
    k_scatter_edges<<<gEw, T, 0, stream>>>(src, dst, h, dinv, agg);

    // Layer 2: h2 = relu(agg1 + b1) @ W2 ; agg2 = scatter(h2) + b2
    k_matmul_wmma<<<gMM, T, 0, stream>>>(agg, W2, b1, h);
    k_init_out<<<gNF, T, 0, stream>>>(h, dinv, b2, agg);
    k_scatter_edges<<<gEw, T, 0, stream>>>(src, dst, h, dinv, agg);

    // Decode
    k_decode<<<gPw, T, 0, stream>>>(pa, pb, agg, out);
}